// SpikeLayer_88553635709313
// MI455X (gfx1250) — compile-verified
//
#include <hip/hip_runtime.h>

// SpikeLayer scan: [T=16, B=512, D=8192] f32.
// V += x[t]; V = (|V| > 1) ? 0 : V; out[t] = V.
// Memory-bound: 512 MB once-through traffic -> ~22 us floor @ 23.3 TB/s.
// Strategy: 1 thread = 4 consecutive features of one chain.
//   - 16x hand-issued global_load_b128 (GVS saddr + scale_offset, th:TH_LOAD_NT)
//     back-to-back: LOADcnt=16 in flight per wave.
//   - one explicit s_wait_loadcnt 0 with all staged regs as tied operands, so
//     no consumer can be scheduled before the data has landed.
//   - serial 16-step integrate-and-reset in registers (the only dependence).
//   - 16x NT global_store_b128 (write-once data).
// No LDS, no WMMA (no contraction exists in this op).

#define TIMES 16
#define SPIKE_THRESHOLD 1.0f

typedef __attribute__((ext_vector_type(4))) float v4f;

__global__ __launch_bounds__(256) void spike_scan_b128(
    const v4f* __restrict__ in,   // [T * n4] float4-view of inputs
    v4f* __restrict__ out,        // same shape
    int n4)                       // B*D/4 = float4 elements per timestep
{
    const int idx = blockIdx.x * blockDim.x + threadIdx.x;
    if (idx >= n4) return;

    // Stage all 16 timesteps with hand-issued B128 NT loads.
    // GVS mode: uniform SGPR-pair base per timestep + per-lane element index
    // scaled by the 16B data size (scale_offset) -- same encoding the compiler
    // emitted for timestep 0 in the previous round.
    v4f x[TIMES];
#pragma unroll
    for (int t = 0; t < TIMES; ++t) {
        const v4f* base_t = in + (size_t)t * (size_t)n4;   // uniform -> SGPRs
        asm volatile("global_load_b128 %0, %1, %2 scale_offset th:TH_LOAD_NT"
                     : "=v"(x[t])
                     : "v"(idx), "s"(base_t));
    }

    // Single wait for all 16 outstanding loads. The "+v" ties make every x[t]
    // data-dependent on this instruction, so the scheduler cannot hoist any
    // consumer above it (the compiler does not track asm loads in LOADcnt).
    asm volatile("s_wait_loadcnt 0x0"
                 : "+v"(x[0]),  "+v"(x[1]),  "+v"(x[2]),  "+v"(x[3]),
                   "+v"(x[4]),  "+v"(x[5]),  "+v"(x[6]),  "+v"(x[7]),
                   "+v"(x[8]),  "+v"(x[9]),  "+v"(x[10]), "+v"(x[11]),
                   "+v"(x[12]), "+v"(x[13]), "+v"(x[14]), "+v"(x[15])
                 :
                 : "memory");

    // Serial integrate-and-reset scan.
    v4f V;
    V.x = 0.0f; V.y = 0.0f; V.z = 0.0f; V.w = 0.0f;
#pragma unroll
    for (int t = 0; t < TIMES; ++t) {
        V.x += x[t].x;
        V.y += x[t].y;
        V.z += x[t].z;
        V.w += x[t].w;
        // |V| > thr -> reset (strict >; NaN compares false -> NaN kept,
        // matching jnp.where semantics).
        V.x = (__builtin_fabsf(V.x) > SPIKE_THRESHOLD) ? 0.0f : V.x;
        V.y = (__builtin_fabsf(V.y) > SPIKE_THRESHOLD) ? 0.0f : V.y;
        V.z = (__builtin_fabsf(V.z) > SPIKE_THRESHOLD) ? 0.0f : V.z;
        V.w = (__builtin_fabsf(V.w) > SPIKE_THRESHOLD) ? 0.0f : V.w;
        x[t] = V;  // reuse staging registers as the output tile
    }

    // Stream results out with NT stores (never re-read). STOREcnt is tracked
    // by the compiler; s_endpgm's implicit wait-idle flushes them.
#pragma unroll
    for (int t = 0; t < TIMES; ++t) {
        v4f* base_t = out + (size_t)t * (size_t)n4;        // uniform -> SGPRs
        __builtin_nontemporal_store(x[t], base_t + idx);
    }
}

extern "C" void kernel_launch(void* const* d_in, const int* in_sizes, int n_in,
                              void* d_out, int out_size, void* d_ws, size_t ws_size,
                              hipStream_t stream) {
    const float* in = (const float*)d_in[0];
    float* out = (float*)d_out;

    const int total    = in_sizes[0];        // T*B*D = 67,108,864
    const int per_step = total / TIMES;      // B*D   = 4,194,304
    const int n4       = per_step / 4;       // float4 per timestep = 1,048,576

    const int block = 256;                   // 8 wave32 waves
    const int grid  = (n4 + block - 1) / block;

    spike_scan_b128<<<grid, block, 0, stream>>>(
        (const v4f*)in, (v4f*)out, n4);
}